// GCNFeatureExtractor_81475529605941
// MI455X (gfx1250) — compile-verified
//
#include <hip/hip_runtime.h>
#include <hip/hip_bf16.h>

typedef __attribute__((ext_vector_type(2))) float v2f;
typedef __attribute__((ext_vector_type(8))) float v8f;

#define NPTS   4096
#define NBATCH 8
#define NNEIGH 8
#define MTOT   (NBATCH * NPTS)
#define SLOPE  0.2f

// ---------------------------------------------------------------------------
// 1) KNN: one block = 256 nodes of one batch; whole batch's points in LDS.
//    Register insertion-sort keeps top-8 smallest distances (incl. self).
// ---------------------------------------------------------------------------
__global__ __launch_bounds__(256) void knn_kernel(const float2* __restrict__ x,
                                                  int* __restrict__ idx)
{
    __shared__ float2 spts[NPTS];                       // 32 KB
    const int b   = blockIdx.y;
    const int tid = threadIdx.x;
    for (int i = tid; i < NPTS; i += 256) spts[i] = x[b * NPTS + i];
    __syncthreads();

    const int    m  = blockIdx.x * 256 + tid;
    const float2 pm = spts[m];

    float bd[NNEIGH];
    int   bi[NNEIGH];
#pragma unroll
    for (int j = 0; j < NNEIGH; ++j) { bd[j] = 3.4e38f; bi[j] = 0; }

    for (int n = 0; n < NPTS; ++n) {
        const float dx = pm.x - spts[n].x;
        const float dy = pm.y - spts[n].y;
        const float d  = dx * dx + dy * dy;
        if (d < bd[NNEIGH - 1]) {
#pragma unroll
            for (int p = NNEIGH - 1; p >= 1; --p) {
                if (d < bd[p - 1])      { bd[p] = bd[p - 1]; bi[p] = bi[p - 1]; }
                else if (d < bd[p])     { bd[p] = d;         bi[p] = n; }
            }
            if (d < bd[0]) { bd[0] = d; bi[0] = n; }
        }
    }
    const int row = b * NPTS + m;
#pragma unroll
    for (int j = 0; j < NNEIGH; ++j) idx[row * NNEIGH + j] = b * NPTS + bi[j];
}

// ---------------------------------------------------------------------------
// 2) Layer 0 (Cin=2): gather-sum 8 neighbors -> tiny 64x2 linear -> h0 (pre-BN)
//    + per-channel sum/sumsq accumulation for BN stats.
//    block (64,4): phase 1 threads gather 256 nodes; phase 2 channel-parallel.
// ---------------------------------------------------------------------------
__global__ __launch_bounds__(256) void layer0_kernel(const float2* __restrict__ x,
                                                     const float*  __restrict__ W0,
                                                     const int*    __restrict__ idx,
                                                     float*        __restrict__ h0,
                                                     float*        __restrict__ ssum,
                                                     float*        __restrict__ ssq)
{
    __shared__ float2 sagg[256];
    __shared__ float  w0s[128];
    const int tx = threadIdx.x, ty = threadIdx.y;
    const int t  = ty * 64 + tx;
    if (t < 128) w0s[t] = W0[t];

    const int  m  = blockIdx.x * 256 + t;
    const int* id = idx + (size_t)m * NNEIGH;
    float2 a; a.x = 0.f; a.y = 0.f;
#pragma unroll
    for (int j = 0; j < NNEIGH; ++j) { const float2 p = x[id[j]]; a.x += p.x; a.y += p.y; }
    sagg[t] = a;
    __syncthreads();

    const float wa = w0s[tx * 2], wb = w0s[tx * 2 + 1];
    const int   base = blockIdx.x * 256;
    float s = 0.f, s2 = 0.f;
    for (int g = 0; g < 64; ++g) {
        const int    j  = g * 4 + ty;
        const float2 ag = sagg[j];
        const float  h  = wa * ag.x + wb * ag.y;
        h0[(size_t)(base + j) * 64 + tx] = h;            // coalesced over tx
        s += h; s2 += h * h;
    }
    atomicAdd(&ssum[tx], s);
    atomicAdd(&ssq[tx],  s2);
}

// ---------------------------------------------------------------------------
// 3) BN finalize: per-channel scale/shift from accumulated sum / sumsq.
// ---------------------------------------------------------------------------
__global__ void bn_finalize_kernel(const float* __restrict__ ssum,
                                   const float* __restrict__ ssq,
                                   const float* __restrict__ g,
                                   const float* __restrict__ bb,
                                   float* __restrict__ scale,
                                   float* __restrict__ shift,
                                   int C, float invM)
{
    const int c = blockIdx.x * blockDim.x + threadIdx.x;
    if (c >= C) return;
    const float mean = ssum[c] * invM;
    const float var  = fmaxf(ssq[c] * invM - mean * mean, 0.f);
    const float sc   = g[c] * rsqrtf(var + 1e-5f);
    scale[c] = sc;
    shift[c] = bb[c] - mean * sc;
}

// ---------------------------------------------------------------------------
// 4) Aggregation for layers 1/2: BN-affine + LeakyReLU applied on the fly
//    while gathering 8 neighbors.  block (C, NPB); rows are C-float contiguous.
// ---------------------------------------------------------------------------
template <int C, int NPB>
__global__ __launch_bounds__(256) void agg_kernel(const float* __restrict__ h,
                                                  const float* __restrict__ scale,
                                                  const float* __restrict__ shift,
                                                  const int*   __restrict__ idx,
                                                  float*       __restrict__ out)
{
    __shared__ int sidx[NPB][NNEIGH];
    const int c = threadIdx.x, ty = threadIdx.y;
    const int m = blockIdx.x * NPB + ty;
    if (c < NNEIGH) sidx[ty][c] = idx[(size_t)m * NNEIGH + c];
    __syncthreads();

    const float sc = scale[c], sh = shift[c];
    float acc = 0.f;
#pragma unroll
    for (int j = 0; j < NNEIGH; ++j) {
        const float v = fmaf(h[(size_t)sidx[ty][j] * C + c], sc, sh);
        acc += (v >= 0.f) ? v : SLOPE * v;
    }
    out[(size_t)m * C + c] = acc;
}

// ---------------------------------------------------------------------------
// 5) fp32 WMMA GEMM:  Cmat[M,Nout] = A[M,K] @ W[Nout,K]^T  (pre-BN output),
//    with fused per-channel sum/sumsq BN-stat accumulation.
//    block (32,8): 8 waves, each wave -> 16(M) x 64(N) tile via 4 accumulators
//    of V_WMMA_F32_16X16X4_F32, A fragment reused across the 4 N-subtiles.
//    ISA f32 16x4 A layout: lanes 0-15 hold K=k0,k0+1; lanes 16-31 K=k0+2,k0+3.
// ---------------------------------------------------------------------------
template <int K>
__global__ __launch_bounds__(256) void gemm_wmma_kernel(const float* __restrict__ A,
                                                        const float* __restrict__ W,
                                                        float*       __restrict__ Cmat,
                                                        float*       __restrict__ ssum,
                                                        float*       __restrict__ ssq,
                                                        int Nout)
{
    __shared__ float lsum[64], lsq[64];
    const int lane = threadIdx.x;
    const int wv   = threadIdx.y;
    const int t    = wv * 32 + lane;
    if (t < 64) { lsum[t] = 0.f; lsq[t] = 0.f; }
    __syncthreads();

    const int half = lane >> 4;          // lane half selects K pair {0,1} vs {2,3}
    const int l15  = lane & 15;          // A row / B column within tile
    const int m0   = (blockIdx.x * 8 + wv) * 16;
    const int n0   = blockIdx.y * 64;

    v8f acc0 = {}, acc1 = {}, acc2 = {}, acc3 = {};

    const float* arow = A + (size_t)(m0 + l15) * K + 2 * half;
    const float* w0p  = W + (size_t)(n0 +  0 + l15) * K + 2 * half;
    const float* w1p  = W + (size_t)(n0 + 16 + l15) * K + 2 * half;
    const float* w2p  = W + (size_t)(n0 + 32 + l15) * K + 2 * half;
    const float* w3p  = W + (size_t)(n0 + 48 + l15) * K + 2 * half;

#pragma unroll
    for (int k0 = 0; k0 < K; k0 += 4) {
        const v2f a  = *(const v2f*)(arow + k0);
        const v2f b0 = *(const v2f*)(w0p + k0);
        const v2f b1 = *(const v2f*)(w1p + k0);
        const v2f b2 = *(const v2f*)(w2p + k0);
        const v2f b3 = *(const v2f*)(w3p + k0);
        acc0 = __builtin_amdgcn_wmma_f32_16x16x4_f32(false, a, false, b0, (short)0, acc0, false, false);
        acc1 = __builtin_amdgcn_wmma_f32_16x16x4_f32(false, a, false, b1, (short)0, acc1, false, false);
        acc2 = __builtin_amdgcn_wmma_f32_16x16x4_f32(false, a, false, b2, (short)0, acc2, false, false);
        acc3 = __builtin_amdgcn_wmma_f32_16x16x4_f32(false, a, false, b3, (short)0, acc3, false, false);
    }

    // Epilogue: C/D layout -> VGPR v holds row m0+v (lanes 0-15) / m0+v+8 (16-31),
    // column = n0 + sub*16 + l15.  Fused BN-stat partial sums via LDS.
#define EPILOG(ACC, SUB)                                                        \
    {                                                                           \
        float s = 0.f, s2 = 0.f;                                                \
        _Pragma("unroll")                                                       \
        for (int v = 0; v < 8; ++v) { const float val = ACC[v]; s += val; s2 += val * val; } \
        atomicAdd(&lsum[SUB * 16 + l15], s);                                    \
        atomicAdd(&lsq [SUB * 16 + l15], s2);                                   \
        const int col = n0 + SUB * 16 + l15;                                    \
        _Pragma("unroll")                                                       \
        for (int v = 0; v < 8; ++v)                                             \
            Cmat[(size_t)(m0 + v + 8 * half) * Nout + col] = ACC[v];            \
    }
    EPILOG(acc0, 0)
    EPILOG(acc1, 1)
    EPILOG(acc2, 2)
    EPILOG(acc3, 3)
#undef EPILOG

    __syncthreads();
    if (t < 64) {
        atomicAdd(&ssum[n0 + t], lsum[t]);
        atomicAdd(&ssq [n0 + t], lsq[t]);
    }
}

// ---------------------------------------------------------------------------
// 6) Final BN + LeakyReLU applied in-place on d_out ([M,1024]), float4 wide.
// ---------------------------------------------------------------------------
__global__ __launch_bounds__(256) void bn_apply_kernel(float* __restrict__ out,
                                                       const float* __restrict__ scale,
                                                       const float* __restrict__ shift)
{
    const size_t i  = (size_t)blockIdx.x * 256 + threadIdx.x;
    const int    c0 = (int)((i * 4) & 1023);
    float4 v = ((float4*)out)[i];
    const float4 sc = *(const float4*)(scale + c0);
    const float4 sh = *(const float4*)(shift + c0);
    float4 y;
    y.x = fmaf(v.x, sc.x, sh.x); y.x = (y.x >= 0.f) ? y.x : SLOPE * y.x;
    y.y = fmaf(v.y, sc.y, sh.y); y.y = (y.y >= 0.f) ? y.y : SLOPE * y.y;
    y.z = fmaf(v.z, sc.z, sh.z); y.z = (y.z >= 0.f) ? y.z : SLOPE * y.z;
    y.w = fmaf(v.w, sc.w, sh.w); y.w = (y.w >= 0.f) ? y.w : SLOPE * y.w;
    ((float4*)out)[i] = y;
}

// ---------------------------------------------------------------------------
extern "C" void kernel_launch(void* const* d_in, const int* in_sizes, int n_in,
                              void* d_out, int out_size, void* d_ws, size_t ws_size,
                              hipStream_t stream)
{
    const float2* x  = (const float2*)d_in[0];
    const float*  W0 = (const float*)d_in[1];
    const float*  W1 = (const float*)d_in[2];
    const float*  W2 = (const float*)d_in[3];
    const float*  g0 = (const float*)d_in[4];
    const float*  b0 = (const float*)d_in[5];
    const float*  g1 = (const float*)d_in[6];
    const float*  b1 = (const float*)d_in[7];
    const float*  g2 = (const float*)d_in[8];
    const float*  b2 = (const float*)d_in[9];
    float* out = (float*)d_out;

    char*  ws    = (char*)d_ws;
    int*   idxb  = (int*)(ws + 0);                  //  1 MiB : [M][8] neighbor rows
    float* h0    = (float*)(ws + (size_t)( 1u << 20)); //  8 MiB : [M][64]  pre-BN
    float* agg1  = (float*)(ws + (size_t)( 9u << 20)); //  8 MiB : [M][64]
    float* h1    = (float*)(ws + (size_t)(17u << 20)); // 16 MiB : [M][128] pre-BN
    float* agg2  = (float*)(ws + (size_t)(33u << 20)); // 16 MiB : [M][128]
    float* ssum  = (float*)(ws + (size_t)(49u << 20)); //  4 KiB
    float* ssq   = ssum  + 1024;                       //  4 KiB
    float* scale = ssq   + 1024;                       //  4 KiB
    float* shift = scale + 1024;                       //  4 KiB

    const float invM = 1.0f / (float)MTOT;

    // graph + layer 0
    knn_kernel<<<dim3(NPTS / 256, NBATCH), 256, 0, stream>>>(x, idxb);
    hipMemsetAsync(ssum, 0, 2 * 1024 * sizeof(float), stream);
    layer0_kernel<<<dim3(MTOT / 256), dim3(64, 4), 0, stream>>>(x, W0, idxb, h0, ssum, ssq);
    bn_finalize_kernel<<<1, 64, 0, stream>>>(ssum, ssq, g0, b0, scale, shift, 64, invM);

    // layer 1: gather(+BN0+lrelu) -> WMMA GEMM (K=64, Nout=128) + stats
    agg_kernel<64, 4><<<dim3(MTOT / 4), dim3(64, 4), 0, stream>>>(h0, scale, shift, idxb, agg1);
    hipMemsetAsync(ssum, 0, 2 * 1024 * sizeof(float), stream);
    gemm_wmma_kernel<64><<<dim3(MTOT / 128, 128 / 64), dim3(32, 8), 0, stream>>>(agg1, W1, h1, ssum, ssq, 128);
    bn_finalize_kernel<<<1, 128, 0, stream>>>(ssum, ssq, g1, b1, scale, shift, 128, invM);

    // layer 2: gather(+BN1+lrelu) -> WMMA GEMM (K=128, Nout=1024) -> d_out + stats
    agg_kernel<128, 2><<<dim3(MTOT / 2), dim3(128, 2), 0, stream>>>(h1, scale, shift, idxb, agg2);
    hipMemsetAsync(ssum, 0, 2 * 1024 * sizeof(float), stream);
    gemm_wmma_kernel<128><<<dim3(MTOT / 128, 1024 / 64), dim3(32, 8), 0, stream>>>(agg2, W2, out, ssum, ssq, 1024);
    bn_finalize_kernel<<<1, 1024, 0, stream>>>(ssum, ssq, g2, b2, scale, shift, 1024, invM);

    // final BN + LeakyReLU in place on the 134 MB output
    bn_apply_kernel<<<dim3((size_t)MTOT * 1024 / 4 / 256), 256, 0, stream>>>(out, scale, shift);
}